// CvxpyProjectionLayer_35648228557558
// MI455X (gfx1250) — compile-verified
//
#include <hip/hip_runtime.h>

// ---------------------------------------------------------------------------
// Batched QP projection via OSQP-style ADMM, MI455X (gfx1250, wave32).
//
// Problem sizes (fixed by reference):
//   BATCH=2048, n=80 actions, ns=4 slacks, m=85 constraints
//   NX = n+ns = 84 (primal dim),  NZ = m+ns+n = 169 (constraint rows)
//
// Design:
//   Kernel 1 (per batch): stage A in LDS, compute AtA with
//     V_WMMA_F32_16X16X4_F32 (5x5 tiles of 16x16, K padded 85->88 with zeros),
//     assemble K = rho*M^T M + diag(p+sigma) analytically (M never formed),
//     invert K with pivotless Gauss-Jordan in LDS (K is SPD), write Kinv to ws.
//   Kernel 2 (per batch): A + Kinv + all vectors resident in LDS (~60KB),
//     150 ADMM iterations; the three matvecs are split 3-ways across the
//     256 threads (partial sums in LDS) for lane utilization.
//
// d_ws requirement: BATCH * 84*85 * 4 bytes = 58.5 MB of scratch for Kinv.
// ---------------------------------------------------------------------------

#define BATCH_C   2048
#define NACT      80
#define NCON      85
#define NSTAB     4
#define NX        84      // NACT + NSTAB
#define NZ        169     // NCON + NSTAB + NACT
#define ITERS_C   150
#define RHO_C     1.0f
#define SIGMA_C   1e-6f
#define ALPHA_C   1.6f
#define BIG_C     1e10f
#define SLACK_C   1.0f    // SLACK_PEN

#define AS_STR    81      // LDS row stride for A (odd -> conflict-free row access)
#define AS_ROWS   88      // K dim padded to multiple of 4 for WMMA
#define AUG_W     168     // [K | I] width
#define AUG_STR   169     // odd stride
#define KV_STR    85      // Kinv LDS/ws row stride (odd)

typedef float v2f __attribute__((ext_vector_type(2)));
typedef float v8f __attribute__((ext_vector_type(8)));

// Check WMMA availability on the DEVICE pass only (host pass never has it).
#if defined(__HIP_DEVICE_COMPILE__)
#if !__has_builtin(__builtin_amdgcn_wmma_f32_16x16x4_f32)
#error "device pass: __builtin_amdgcn_wmma_f32_16x16x4_f32 not available"
#endif
#endif

// ---------------------------------------------------------------------------
// Kernel 1: build K via WMMA AtA, Gauss-Jordan invert, store Kinv to ws.
// One 256-thread (8-wave) workgroup per batch element.
// ---------------------------------------------------------------------------
__global__ __launch_bounds__(256) void setup_invert_kernel(
    const float* __restrict__ A, float* __restrict__ kinv_ws)
{
    const int b   = blockIdx.x;
    const int tid = threadIdx.x;

    __shared__ float As[AS_ROWS * AS_STR];    // 28512 B
    __shared__ float aug[NX * AUG_STR];       // 56784 B  [K | I]
    __shared__ float fcol[NX];
    __shared__ float pivinv;

    const float* Ab = A + (size_t)b * (NCON * NACT);

    // Stage A into LDS (row-major, stride 81); zero pad rows 85..87 + pad col.
    for (int idx = tid; idx < AS_ROWS * AS_STR; idx += 256) {
        int r = idx / AS_STR, c = idx % AS_STR;
        As[idx] = (r < NCON && c < NACT) ? Ab[r * NACT + c] : 0.0f;
    }
    __syncthreads();

    // ---- AtA via V_WMMA_F32_16X16X4_F32 -----------------------------------
    // Output tile (I..I+15, J..J+15) of AtA; one tile per wave, 25 tiles.
    // A-frag (16x4 f32): lanes 0-15 hold M=lane, K=(0,1); lanes 16-31 K=(2,3).
    //   A_frag[m][k] = A[kbase+k][I+m]  (transposed read of A)
    // B-frag (4x16 f32), row-striped across lanes mirroring the K split:
    //   B_frag[k][nn] = A[kbase+k][J+nn]
    // C/D (16x16 f32, 8 VGPRs): VGPR j -> row I+j (lanes 0-15) / I+j+8
    // (lanes 16-31), col J + (lane&15).
    const int wave  = tid >> 5;
    const int lane  = tid & 31;
    const int mrow  = lane & 15;
    const int khalf = (lane < 16) ? 0 : 2;

    for (int t = wave; t < 25; t += 8) {
        const int I = (t / 5) * 16;
        const int J = (t % 5) * 16;
        v8f acc = {0.f, 0.f, 0.f, 0.f, 0.f, 0.f, 0.f, 0.f};
        for (int k = 0; k < AS_ROWS; k += 4) {
#if defined(__HIP_DEVICE_COMPILE__)
            v2f af, bf;
            af.x = As[(k + khalf    ) * AS_STR + I + mrow];
            af.y = As[(k + khalf + 1) * AS_STR + I + mrow];
            bf.x = As[(k + khalf    ) * AS_STR + J + mrow];
            bf.y = As[(k + khalf + 1) * AS_STR + J + mrow];
            acc = __builtin_amdgcn_wmma_f32_16x16x4_f32(
                false, af, false, bf, (short)0, acc, false, false);
#endif
        }
        const int rbase = I + ((lane < 16) ? 0 : 8);
#pragma unroll
        for (int j = 0; j < 8; ++j)
            aug[(rbase + j) * AUG_STR + J + mrow] = acc[j];
    }
    __syncthreads();

    // ---- Assemble augmented [K | I] ---------------------------------------
    // K_xx = rho*(AtA + I) + (1+sigma)I ; K_xs = -rho*A_s^T ; K_ss = (2rho+2+sigma)I
    for (int idx = tid; idx < NX * AUG_W; idx += 256) {
        int i = idx / AUG_W, j = idx % AUG_W;
        float v;
        if (j >= NX) {
            v = (j - NX == i) ? 1.0f : 0.0f;
        } else if (i < NACT && j < NACT) {
            float ata = aug[i * AUG_STR + j];
            v = RHO_C * ata + ((i == j) ? (RHO_C + 1.0f + SIGMA_C) : 0.0f);
        } else if (i < NACT) {              // j in 80..83
            v = -RHO_C * As[(81 + (j - NACT)) * AS_STR + i];
        } else if (j < NACT) {              // i in 80..83
            v = -RHO_C * As[(81 + (i - NACT)) * AS_STR + j];
        } else {
            v = (i == j) ? (2.0f * RHO_C + 2.0f * SLACK_C + SIGMA_C) : 0.0f;
        }
        aug[i * AUG_STR + j] = v;
    }
    __syncthreads();

    // ---- Gauss-Jordan (pivotless; K SPD, diag >= 2) -----------------------
    for (int p = 0; p < NX; ++p) {
        if (tid < NX) fcol[tid] = (tid == p) ? 0.0f : aug[tid * AUG_STR + p];
        if (tid == 0) pivinv = 1.0f / aug[p * AUG_STR + p];
        __syncthreads();
        for (int j = tid; j < AUG_W; j += 256) aug[p * AUG_STR + j] *= pivinv;
        __syncthreads();
        for (int idx = tid; idx < NX * AUG_W; idx += 256) {
            int i = idx / AUG_W, j = idx % AUG_W;
            if (i != p)
                aug[i * AUG_STR + j] -= fcol[i] * aug[p * AUG_STR + j];
        }
        __syncthreads();
    }

    // ---- Write Kinv (stride-85 padded) to workspace -----------------------
    float* Kb = kinv_ws + (size_t)b * (NX * KV_STR);
    for (int idx = tid; idx < NX * KV_STR; idx += 256) {
        int i = idx / KV_STR, j = idx % KV_STR;
        Kb[idx] = (j < NX) ? aug[i * AUG_STR + NX + j] : 0.0f;
    }
}

// ---------------------------------------------------------------------------
// Kernel 2: 150 ADMM iterations, everything in LDS. One 256-thread WG/batch.
// ---------------------------------------------------------------------------
__global__ __launch_bounds__(256) void admm_kernel(
    const float* __restrict__ x_raw, const float* __restrict__ A,
    const float* __restrict__ bcon,  const float* __restrict__ lower,
    const float* __restrict__ upper, const float* __restrict__ kinv_ws,
    float* __restrict__ out)
{
    const int b   = blockIdx.x;
    const int tid = threadIdx.x;

    __shared__ float As[NCON * AS_STR];   // 27540 B
    __shared__ float Kv[NX * KV_STR];     // 28560 B
    __shared__ float xv[NX], xt[NX], rhs[NX];
    __shared__ float zv[NZ], yv[NZ], tv[NZ];
    __shared__ float bb[NCON], lo[NACT], hi[NACT], xr[NACT];
    __shared__ float part[256];           // split-dot partials (max 255 used)

    const float* Ab = A      + (size_t)b * (NCON * NACT);
    const float* Kb = kinv_ws + (size_t)b * (NX * KV_STR);

    for (int idx = tid; idx < NCON * AS_STR; idx += 256) {
        int r = idx / AS_STR, c = idx % AS_STR;
        As[idx] = (c < NACT) ? Ab[r * NACT + c] : 0.0f;
    }
    for (int idx = tid; idx < NX * KV_STR; idx += 256) Kv[idx] = Kb[idx];
    for (int i = tid; i < NCON; i += 256) bb[i] = bcon[(size_t)b * NCON + i];
    for (int i = tid; i < NACT; i += 256) {
        lo[i] = lower[(size_t)b * NACT + i];
        hi[i] = upper[(size_t)b * NACT + i];
        xr[i] = x_raw[(size_t)b * NACT + i];
    }
    if (tid < NX)  xv[tid] = (tid < NACT) ? x_raw[(size_t)b * NACT + tid] : 0.0f;
    for (int r = tid; r < NZ; r += 256) yv[r] = 0.0f;
    __syncthreads();

    // z0 = M x0   (rows 0..84: A*x0; 81..84 also -s but s0=0; 85..88: s0; 89..: x0)
    if (tid < 255) {
        int pt = tid / NCON, r = tid % NCON;
        int j0 = (pt * NACT) / 3, j1 = ((pt + 1) * NACT) / 3;
        float s = 0.0f;
        for (int j = j0; j < j1; ++j) s += As[r * AS_STR + j] * xv[j];
        part[pt * NCON + r] = s;
    }
    __syncthreads();
    for (int r = tid; r < NZ; r += 256) {
        float v;
        if (r < NCON) {
            v = part[r] + part[NCON + r] + part[2 * NCON + r];
            if (r >= 81) v -= xv[NACT + (r - 81)];
        } else if (r < NCON + NSTAB) {
            v = xv[NACT + (r - NCON)];
        } else {
            v = xv[r - (NCON + NSTAB)];
        }
        zv[r] = v;
    }
    __syncthreads();

    for (int it = 0; it < ITERS_C; ++it) {
        // ph1: t = rho*z - y
        for (int r = tid; r < NZ; r += 256) tv[r] = RHO_C * zv[r] - yv[r];
        __syncthreads();

        // ph2: partials of A^T t (3 x 80 threads); slack rhs on threads 240..243
        if (tid < 240) {
            int pt = tid / NACT, i = tid % NACT;
            int r0 = (pt * NCON) / 3, r1 = ((pt + 1) * NCON) / 3;
            float s = 0.0f;
            for (int r = r0; r < r1; ++r) s += As[r * AS_STR + i] * tv[r];
            part[pt * NACT + i] = s;
        } else if (tid < 240 + NSTAB) {
            int j = tid - 240;
            rhs[NACT + j] = SIGMA_C * xv[NACT + j] - tv[81 + j] + tv[NCON + j];
        }
        __syncthreads();

        // ph3: rhs (x part) = sigma*x + x_raw + t[89+i] + A^T t
        if (tid < NACT)
            rhs[tid] = SIGMA_C * xv[tid] + xr[tid] + tv[NCON + NSTAB + tid]
                     + part[tid] + part[NACT + tid] + part[2 * NACT + tid];
        __syncthreads();

        // ph4: xt = Kinv * rhs, split 3 x 84 (j-chunks of 28)
        if (tid < 252) {
            int pt = tid / NX, i = tid % NX;
            int j0 = pt * 28, j1 = j0 + 28;
            float s = 0.0f;
            for (int j = j0; j < j1; ++j) s += Kv[i * KV_STR + j] * rhs[j];
            part[pt * NX + i] = s;
        }
        __syncthreads();
        if (tid < NX) xt[tid] = part[tid] + part[NX + tid] + part[2 * NX + tid];
        __syncthreads();

        // ph6: A * xt row partials (3 x 85)
        if (tid < 255) {
            int pt = tid / NCON, r = tid % NCON;
            int j0 = (pt * NACT) / 3, j1 = ((pt + 1) * NACT) / 3;
            float s = 0.0f;
            for (int j = j0; j < j1; ++j) s += As[r * AS_STR + j] * xt[j];
            part[pt * NCON + r] = s;
        }
        __syncthreads();

        // ph7: zt = M*xt; relaxed z; clip; dual update. x update on spare lanes.
        for (int r = tid; r < NZ; r += 256) {
            float ztr;
            if (r < NCON) {
                ztr = part[r] + part[NCON + r] + part[2 * NCON + r];
                if (r >= 81) ztr -= xt[NACT + (r - 81)];
            } else if (r < NCON + NSTAB) {
                ztr = xt[NACT + (r - NCON)];
            } else {
                ztr = xt[r - (NCON + NSTAB)];
            }
            float zr = ALPHA_C * ztr + (1.0f - ALPHA_C) * zv[r];
            float lc, uc;
            if (r < NCON)            { lc = -BIG_C;    uc = bb[r]; }
            else if (r < NCON+NSTAB) { lc = 0.0f;      uc = BIG_C; }
            else                     { lc = lo[r-(NCON+NSTAB)]; uc = hi[r-(NCON+NSTAB)]; }
            float v  = zr + yv[r] / RHO_C;
            float zn = fminf(fmaxf(v, lc), uc);
            yv[r] = yv[r] + RHO_C * (zr - zn);
            zv[r] = zn;
        }
        if (tid >= NZ && tid < NZ + NX) {   // threads 169..252: x update
            int i = tid - NZ;
            xv[i] = ALPHA_C * xt[i] + (1.0f - ALPHA_C) * xv[i];
        }
        __syncthreads();
    }

    if (tid < NACT) out[(size_t)b * NACT + tid] = xv[tid];
}

extern "C" void kernel_launch(void* const* d_in, const int* in_sizes, int n_in,
                              void* d_out, int out_size, void* d_ws, size_t ws_size,
                              hipStream_t stream) {
    (void)in_sizes; (void)n_in; (void)out_size; (void)ws_size;
    const float* x_raw = (const float*)d_in[0];
    const float* A     = (const float*)d_in[1];
    const float* bcon  = (const float*)d_in[2];
    const float* lower = (const float*)d_in[3];
    const float* upper = (const float*)d_in[4];
    float* kinv = (float*)d_ws;   // needs BATCH*84*85*4 = 58.5 MB scratch

    setup_invert_kernel<<<dim3(BATCH_C), dim3(256), 0, stream>>>(A, kinv);
    admm_kernel<<<dim3(BATCH_C), dim3(256), 0, stream>>>(
        x_raw, A, bcon, lower, upper, kinv, (float*)d_out);
}